// EmbeddingModel_65274912965363
// MI455X (gfx1250) — compile-verified
//
#include <hip/hip_runtime.h>
#include <stdint.h>

#define B_N   8192
#define NM    64
#define TPS   16
#define LEAF  64
#define EMB_D 32
#define FEAT  (NM * EMB_D)   /* 2048 */
#define XCOLS (NM * TPS)     /* 1024 */
#define ROWS1 512
#define CH1   (B_N / ROWS1)  /* 16 chunks */
#define BN_EPS 1e-5f

typedef float v2f __attribute__((ext_vector_type(2)));
typedef float v8f __attribute__((ext_vector_type(8)));
typedef int   v4i __attribute__((ext_vector_type(4)));

typedef v4i __attribute__((address_space(1))) gv4i;   /* global 128b chunk */
typedef v4i __attribute__((address_space(3))) lv4i;   /* LDS    128b chunk */

#if defined(__has_builtin)
#if __has_builtin(__builtin_amdgcn_global_load_async_to_lds_b128) && \
    __has_builtin(__builtin_amdgcn_s_wait_asynccnt)
#define USE_ASYNC_LDS 1
#endif
#endif

// ---------------------------------------------------------------------------
// Kernel 1: per (model, 512-row chunk): stage 128KB table in LDS via async
// global->LDS b128 copies (ASYNCcnt-tracked), gather+sum 16 embeddings per
// (b,m), write h, emit deterministic BN partial sums.
// ---------------------------------------------------------------------------
__global__ __launch_bounds__(256) void k1_gather(
    const int* __restrict__ x, const float* __restrict__ embw,
    float* __restrict__ h, float* __restrict__ psum, float* __restrict__ psumsq)
{
    __shared__ __align__(16) float tbl[TPS * LEAF * EMB_D];   // 131072 bytes
    __shared__ float red[2][8][EMB_D];

    const int m   = blockIdx.y;
    const int b0  = blockIdx.x * ROWS1;
    const int tid = threadIdx.x;

    const float* tw = embw + (size_t)m * (TPS * LEAF * EMB_D);
#ifdef USE_ASYNC_LDS
    // Async DMA-style staging: 16B per lane per step, tracked by ASYNCcnt.
    for (int i = tid * 4; i < TPS * LEAF * EMB_D; i += 256 * 4) {
        gv4i* gsrc = (gv4i*)(uintptr_t)(tw + i);
        // flat LDS address low 32 bits == LDS byte offset (aperture mapping)
        lv4i* ldst = (lv4i*)(uint32_t)(uintptr_t)(tbl + i);
        __builtin_amdgcn_global_load_async_to_lds_b128(gsrc, ldst, 0, 0);
    }
    __builtin_amdgcn_s_wait_asynccnt(0);
#else
    for (int i = tid; i < TPS * LEAF * EMB_D; i += 256) tbl[i] = tw[i];
#endif
    __syncthreads();

    const int wave = tid >> 5;
    const int lane = tid & 31;          // lane == embedding dim e
    const int rpw  = ROWS1 / 8;         // 64 rows per wave

    float s = 0.f, ss = 0.f;
    for (int r = 0; r < rpw; ++r) {
        const int b  = b0 + wave * rpw + r;
        // lanes 0..15 fetch the 16 table indices for this (b,m); broadcast via shfl
        const int xi = x[(size_t)b * XCOLS + m * TPS + (lane & 15)];
        float acc = 0.f;
#pragma unroll
        for (int t = 0; t < TPS; ++t) {
            const int idx = __shfl(xi, t, 32);
            acc += tbl[(t * LEAF + idx) * EMB_D + lane];   // conflict-free: 32 consecutive banks
        }
        h[(size_t)b * FEAT + m * EMB_D + lane] = acc;
        s  += acc;
        ss += acc * acc;
    }
    red[0][wave][lane] = s;
    red[1][wave][lane] = ss;
    __syncthreads();
    if (wave == 0) {
        float a = 0.f, q = 0.f;
#pragma unroll
        for (int w2 = 0; w2 < 8; ++w2) { a += red[0][w2][lane]; q += red[1][w2][lane]; }
        psum  [((size_t)m * CH1 + blockIdx.x) * EMB_D + lane] = a;
        psumsq[((size_t)m * CH1 + blockIdx.x) * EMB_D + lane] = q;
    }
}

// ---------------------------------------------------------------------------
// Kernel 2: finalize BN stats, fold BN + output weights:
//   alpha[m,e] = gamma*rsqrt(var+eps)*w ;  c[m] = b_out + sum_e (beta-mean*g)*w
// ---------------------------------------------------------------------------
__global__ __launch_bounds__(32) void k2_stats(
    const float* __restrict__ psum, const float* __restrict__ psumsq,
    const float* __restrict__ gamma, const float* __restrict__ beta,
    const float* __restrict__ bw, const float* __restrict__ bb,
    float* __restrict__ alpha, float* __restrict__ cvec)
{
    const int m = blockIdx.x;
    const int e = threadIdx.x;
    float s = 0.f, q = 0.f;
    for (int c = 0; c < CH1; ++c) {
        s += psum  [((size_t)m * CH1 + c) * EMB_D + e];
        q += psumsq[((size_t)m * CH1 + c) * EMB_D + e];
    }
    const float mean = s * (1.0f / B_N);
    const float var  = q * (1.0f / B_N) - mean * mean;
    const float g    = gamma[m * EMB_D + e] * rsqrtf(var + BN_EPS);
    const float w    = bw[m * EMB_D + e];
    alpha[m * EMB_D + e] = g * w;
    float ct = (beta[m * EMB_D + e] - mean * g) * w;
#pragma unroll
    for (int off = 16; off > 0; off >>= 1) ct += __shfl_down(ct, off, 32);
    if (e == 0) cvec[m] = ct + bb[m];
}

// ---------------------------------------------------------------------------
// Kernel 3: out[b,m] = sum_e h[b,m,e]*alpha[m,e] + c[m] via fp32 WMMA chains.
// Block = 16 batch rows x 64 models (8 waves x 8 models). Per model: hoist all
// 16 A/B fragment loads, then a back-to-back chain of 8x V_WMMA_F32_16X16X4_F32
// over E=32 (B-matrix columns replicated with alpha). Output tile staged in
// padded LDS -> coalesced float4 stores; row sums reduced deterministically.
// ---------------------------------------------------------------------------
#define TP 65   /* LDS tile row pitch (padded, conflict-free) */

__global__ __launch_bounds__(256) void k3_out(
    const float* __restrict__ h, const float* __restrict__ alpha,
    const float* __restrict__ cvec, float* __restrict__ sum_out,
    float* __restrict__ out)
{
    __shared__ float tile[16 * TP];
    __shared__ float rs[8][16];
    const int tid  = threadIdx.x;
    const int wave = tid >> 5;
    const int lane = tid & 31;
    const int b0   = blockIdx.x * 16;
    const int row  = lane & 15;   // A: M index ; C/D: N index
    const int hi   = lane >> 4;   // lane-half selects K pair / M+8 block

    float rsum[8];
#pragma unroll
    for (int r = 0; r < 8; ++r) rsum[r] = 0.f;

    for (int j = 0; j < 8; ++j) {
        const int m = wave * 8 + j;
        // A 16x4 f32: lanes 0-15 hold K={e0,e0+1}, lanes 16-31 K={e0+2,e0+3}
        const v2f* hp2 = (const v2f*)(h + (size_t)(b0 + row) * FEAT + m * EMB_D + 2 * hi);
        // B 4x16 f32: all 16 N columns replicated (values depend on K only)
        const v2f* ap2 = (const v2f*)(alpha + m * EMB_D + 2 * hi);

        v2f Av[8], Bv[8];
#pragma unroll
        for (int kc = 0; kc < 8; ++kc) {
            Av[kc] = hp2[2 * kc];
            Bv[kc] = ap2[2 * kc];
        }
        v8f d = {};
#pragma unroll
        for (int kc = 0; kc < 8; ++kc) {
            d = __builtin_amdgcn_wmma_f32_16x16x4_f32(
                    false, Av[kc], false, Bv[kc], (short)0, d, false, false);
        }
        const float cm = cvec[m];
        if (row == 0) {
#pragma unroll
            for (int r = 0; r < 8; ++r) {
                // D layout: VGPR r, lane half hi -> element out[b0 + r + 8*hi, N]
                const float o = d[r] + cm;
                rsum[r] += o;
                tile[(8 * hi + r) * TP + m] = o;
            }
        }
    }
    if (row == 0) {
#pragma unroll
        for (int r = 0; r < 8; ++r) rs[wave][8 * hi + r] = rsum[r];
    }
    __syncthreads();

    // Coalesced output tile store: 256 threads x float4 covers 16x64.
    {
        const int rr = tid >> 4;
        const int c0 = (tid & 15) * 4;
        float4 o;
        o.x = tile[rr * TP + c0 + 0];
        o.y = tile[rr * TP + c0 + 1];
        o.z = tile[rr * TP + c0 + 2];
        o.w = tile[rr * TP + c0 + 3];
        *(float4*)(out + (size_t)(b0 + rr) * NM + c0) = o;
    }
    if (tid < 16) {
        float s = 0.f;
#pragma unroll
        for (int w2 = 0; w2 < 8; ++w2) s += rs[w2][tid];
        sum_out[b0 + tid] = s;
    }
}

// ---------------------------------------------------------------------------
extern "C" void kernel_launch(void* const* d_in, const int* in_sizes, int n_in,
                              void* d_out, int out_size, void* d_ws, size_t ws_size,
                              hipStream_t stream)
{
    const int*   x     = (const int*)  d_in[0];
    const float* embw  = (const float*)d_in[1];
    const float* gamma = (const float*)d_in[2];
    const float* beta  = (const float*)d_in[3];
    const float* bw    = (const float*)d_in[4];
    const float* bb    = (const float*)d_in[5];

    float* sum_out = (float*)d_out;          // (B,1) first
    float* out     = (float*)d_out + B_N;    // then (B, NM)

    float* ws     = (float*)d_ws;
    float* h      = ws;                                        // B*2048 f32 (64 MB)
    float* psum   = h      + (size_t)B_N * FEAT;
    float* psumsq = psum   + (size_t)NM * CH1 * EMB_D;
    float* alpha  = psumsq + (size_t)NM * CH1 * EMB_D;
    float* cvec   = alpha  + FEAT;

    dim3 g1(CH1, NM);
    k1_gather<<<g1, 256, 0, stream>>>(x, embw, h, psum, psumsq);
    k2_stats<<<NM, 32, 0, stream>>>(psum, psumsq, gamma, beta, bw, bb, alpha, cvec);
    k3_out<<<B_N / 16, 256, 0, stream>>>(h, alpha, cvec, sum_out, out);
}